// Nonlinearity_35218731828137
// MI455X (gfx1250) — compile-verified
//
#include <hip/hip_runtime.h>

typedef __attribute__((ext_vector_type(2))) float v2f;
typedef __attribute__((ext_vector_type(8))) float v8f;

#define BDIM 512
#define CDIM 512
#define NROWS (BDIM * CDIM)

// flat-element offsets of out0..out5 in the concatenated output tuple
__device__ __forceinline__ int outOffOf(int L) {
  return L == 0 ? 0
       : L == 1 ? 1572864
       : L == 2 ? 5505024
       : L == 3 ? 12058624
       : L == 4 ? 19398656
       :          28835840;
}

// One kernel per degree l1.  D = A x B + C with
//   A[16 x 4]  : 16 (b,i) rows of pair-products P[bi,k] = x[bi,k/W]*x[bi,k%W]
//   B[4 x 16]  : swizzled CG constants from LDS
//   D[16 x 16] : 16 rows x 16 output columns (col -> (L, M) = (isqrt, col-L^2))
template <int L1>
__global__ __launch_bounds__(256) void tp_kernel(
    const float* __restrict__ x,
    const float* __restrict__ cg0, const float* __restrict__ cg1,
    const float* __restrict__ cg2, const float* __restrict__ cg3,
    const float* __restrict__ cg4, const float* __restrict__ cg5,
    float* __restrict__ out) {
  constexpr int W     = 2 * L1 + 1;
  constexpr int K     = W * W;
  constexpr int LMAX  = (2 * L1 < 5) ? 2 * L1 : 5;
  constexpr int MOUT  = (LMAX + 1) * (LMAX + 1);
  constexpr int STEPS = (K + 3) / 4;          // K padded to multiple of 4
  constexpr int MT    = (MOUT + 15) / 16;     // 16-wide output column tiles

  __shared__ float ldsB[MT * STEPS * 64];     // per (tile,step): 32 lanes x 2 f32

  const int lane = threadIdx.x & 31;
  const int wave = threadIdx.x >> 5;
  const int hsel = (lane >> 4) & 1;           // half-wave select (K offset +2)
  const int l15  = lane & 15;

  // prefetch this wave's x rows while the block builds the CG operand in LDS
  {
    const int pr = (blockIdx.x * 8 + wave) * 16 + l15;
    if (pr < NROWS) __builtin_prefetch(x + (long)pr * W, 0, 0);
  }

  // ---- cooperative fill of the swizzled CG (B-operand) image in LDS ----
  // B 4x16 lane layout: VGPR0 = CG[k0+2*half][col], VGPR1 = CG[k0+1+2*half][col]
  for (int e = threadIdx.x; e < MT * STEPS * 32; e += 256) {
    const int t    = e / (STEPS * 32);
    const int r    = e - t * (STEPS * 32);
    const int s    = r >> 5;
    const int slot = r & 31;
    const int col  = t * 16 + (slot & 15);
    const int k0   = 4 * s + 2 * (slot >> 4);
    float v0 = 0.f, v1 = 0.f;
    if (col < MOUT) {
      const int L  = col >= 25 ? 5 : col >= 16 ? 4 : col >= 9 ? 3
                   : col >= 4  ? 2 : col >= 1  ? 1 : 0;
      const int Ml = col - L * L;
      const float* cgl = L == 0 ? cg0 : L == 1 ? cg1 : L == 2 ? cg2
                       : L == 3 ? cg3 : L == 4 ? cg4 : cg5;
      if (k0 < K)     { int n = k0 / W;      int m = k0 - n * W;      v0 = cgl[(Ml * W + n) * W + m]; }
      if (k0 + 1 < K) { int n = (k0+1) / W;  int m = (k0+1) - n * W;  v1 = cgl[(Ml * W + n) * W + m]; }
    }
    ldsB[2 * e]     = v0;
    ldsB[2 * e + 1] = v1;
  }
  __syncthreads();

  // ---- per-column store constants (depend only on lane) ----
  int  cOff[MT], sB[MT], sI[MT];
  bool ok[MT];
#pragma unroll
  for (int t = 0; t < MT; ++t) {
    const int col = t * 16 + l15;
    ok[t] = (col < MOUT);
    const int cc    = ok[t] ? col : 0;
    const int L     = cc >= 25 ? 5 : cc >= 16 ? 4 : cc >= 9 ? 3
                    : cc >= 4  ? 2 : cc >= 1  ? 1 : 0;
    const int Ml    = cc - L * L;
    const int l1min = (L + 1) >> 1;           // smallest l1 contributing to L
    const int t1    = 2 * L + 1;
    sB[t]   = (6 - l1min) * CDIM * t1;        // stride of b in out_L
    sI[t]   = t1;                             // stride of i in out_L
    cOff[t] = outOffOf(L) + (L1 - l1min) * CDIM * t1 + Ml;
  }

  // ---- main loop: one 16-row tile per wave ----
  const int tile0 = blockIdx.x * 8 + wave;
  for (int tile = tile0; tile < NROWS / 16; tile += gridDim.x * 8) {
    const int rowBase = tile * 16;
    const float* xrow = x + (long)(rowBase + l15) * W;
    float xv[W];
#pragma unroll
    for (int q = 0; q < W; ++q) xv[q] = xrow[q];

#pragma unroll
    for (int t = 0; t < MT; ++t) {
      v8f acc = {};
#pragma unroll
      for (int s = 0; s < STEPS; ++s) {
        const int k0 = 4 * s;
        // pair products for k0..k0+3 (compile-time register indices; 0 pads K)
        const float p0a = (k0     < K) ? xv[(k0    ) / W] * xv[(k0    ) % W] : 0.f;
        const float p1a = (k0 + 1 < K) ? xv[(k0 + 1) / W] * xv[(k0 + 1) % W] : 0.f;
        const float p0b = (k0 + 2 < K) ? xv[(k0 + 2) / W] * xv[(k0 + 2) % W] : 0.f;
        const float p1b = (k0 + 3 < K) ? xv[(k0 + 3) / W] * xv[(k0 + 3) % W] : 0.f;
        v2f a;
        a.x = hsel ? p0b : p0a;
        a.y = hsel ? p1b : p1a;
        const v2f b = *(const v2f*)(ldsB + ((t * STEPS + s) * 32 + lane) * 2);
        acc = __builtin_amdgcn_wmma_f32_16x16x4_f32(
            /*neg_a=*/false, a, /*neg_b=*/false, b,
            /*c_mod=*/(short)0, acc, /*reuse_a=*/false, /*reuse_b=*/false);
      }
      if (ok[t]) {
#pragma unroll
        for (int g = 0; g < 8; ++g) {
          const int  bi = rowBase + g + 8 * hsel;  // D row for this VGPR/half
          const int  bb = bi >> 9;                 // C == 512
          const int  ii = bi & (CDIM - 1);
          const long ad = (long)cOff[t] + (long)bb * sB[t] + (long)ii * sI[t];
          out[ad] = acc[g];
        }
      }
    }
  }
}

extern "C" void kernel_launch(void* const* d_in, const int* in_sizes, int n_in,
                              void* d_out, int out_size, void* d_ws, size_t ws_size,
                              hipStream_t stream) {
  (void)in_sizes; (void)n_in; (void)out_size; (void)d_ws; (void)ws_size;
  const float* x[6];
  for (int l = 0; l < 6; ++l) x[l] = (const float*)d_in[l];
  const float* cg[6][6] = {};          // cg[l1][L], unused stay nullptr
  int idx = 6;
  for (int l1 = 0; l1 < 6; ++l1) {
    const int lmax = (2 * l1 < 5) ? 2 * l1 : 5;
    for (int L = 0; L <= lmax; ++L) cg[l1][L] = (const float*)d_in[idx++];
  }
  float* out = (float*)d_out;

  dim3 grid(2048), block(256);   // 2048*8 waves == 16384 row tiles (one each)
  tp_kernel<0><<<grid, block, 0, stream>>>(x[0], cg[0][0], cg[0][1], cg[0][2], cg[0][3], cg[0][4], cg[0][5], out);
  tp_kernel<1><<<grid, block, 0, stream>>>(x[1], cg[1][0], cg[1][1], cg[1][2], cg[1][3], cg[1][4], cg[1][5], out);
  tp_kernel<2><<<grid, block, 0, stream>>>(x[2], cg[2][0], cg[2][1], cg[2][2], cg[2][3], cg[2][4], cg[2][5], out);
  tp_kernel<3><<<grid, block, 0, stream>>>(x[3], cg[3][0], cg[3][1], cg[3][2], cg[3][3], cg[3][4], cg[3][5], out);
  tp_kernel<4><<<grid, block, 0, stream>>>(x[4], cg[4][0], cg[4][1], cg[4][2], cg[4][3], cg[4][4], cg[4][5], out);
  tp_kernel<5><<<grid, block, 0, stream>>>(x[5], cg[5][0], cg[5][1], cg[5][2], cg[5][3], cg[5][4], cg[5][5], out);
}